// GeometricCorrelationMap_91207925498299
// MI455X (gfx1250) — compile-verified
//
#include <hip/hip_runtime.h>
#include <hip/hip_bf16.h>

// Problem constants (from reference: [8, 256, 48, 48])
#define BATCH 8
#define CDIM  256
#define SDIM  2304            // 48*48

typedef __attribute__((ext_vector_type(2))) float v2f;
typedef __attribute__((ext_vector_type(8))) float v8f;

// ---------------------------------------------------------------------------
// Kernel 1: inverse L2 norms along C for both src and dst.
// idx in [0, 2*B*S): first half -> src norms, second half -> dst norms.
// Layout in d_ws: inv[0 .. B*S) = src, inv[B*S .. 2*B*S) = dst.
// ---------------------------------------------------------------------------
__global__ __launch_bounds__(256)
void gcm_norms_kernel(const float* __restrict__ src,
                      const float* __restrict__ dst,
                      float* __restrict__ inv) {
    const int total = BATCH * SDIM;
    int idx = blockIdx.x * blockDim.x + threadIdx.x;
    if (idx >= 2 * total) return;
    const float* p = (idx < total) ? src : dst;
    int i = (idx < total) ? idx : (idx - total);
    int b = i / SDIM;
    int s = i - b * SDIM;
    const float* base = p + (size_t)b * CDIM * SDIM + s;
    float acc = 0.0f;
#pragma unroll 8
    for (int c = 0; c < CDIM; ++c) {
        float v = base[(size_t)c * SDIM];
        acc = fmaf(v, v, acc);
    }
    inv[idx] = 1.0f / sqrtf(acc);
}

// ---------------------------------------------------------------------------
// Kernel 2: fused normalized-GEMM + ReLU using V_WMMA_F32_16X16X4_F32.
//   gcm[b, m, n] = relu( (sum_c src[b,c,m] * dst[b,c,n]) * isrc[b,m] * idst[b,n] )
// Both operands are K-major in memory (stride S per c), matching the WMMA f32
// fragment layout:
//   A frag (16x4):  lane L, vgpr j -> (M = L%16, K = 2*(L>>4) + j)
//   B frag (4x16):  lane L, vgpr j -> (N = L%16, K = 2*(L>>4) + j)
//   C/D (16x16):    lane L, vgpr r -> (M = r + 8*(L>>4), N = L%16)
// Block: 256 threads = 8 waves; wave grid 4(M) x 2(N); wave tile 32x64.
// Block tile: 128 x 128. Grid: (18, 18, 8).
// ---------------------------------------------------------------------------
__global__ __launch_bounds__(256)
void gcm_wmma_kernel(const float* __restrict__ src,
                     const float* __restrict__ dst,
                     const float* __restrict__ inv_src,
                     const float* __restrict__ inv_dst,
                     float* __restrict__ out) {
    const int lane = threadIdx.x & 31;
    const int wave = threadIdx.x >> 5;
    const int wm   = wave >> 1;          // 0..3
    const int wn   = wave & 1;           // 0..1
    const int b    = blockIdx.z;

    const int m0 = blockIdx.y * 128 + wm * 32;   // wave M origin (2 x 16)
    const int n0 = blockIdx.x * 128 + wn * 64;   // wave N origin (4 x 16)

    const int lmod = lane & 15;
    const int lhi  = lane >> 4;                  // 0 or 1 (K half-wave)

    const size_t batch_off = (size_t)b * CDIM * SDIM;
    // lane-private base pointers at k = 0 (j=0 element); j=1 is +SDIM floats
    const float* Abase = src + batch_off + (size_t)(lhi * 2) * SDIM + (m0 + lmod);
    const float* Bbase = dst + batch_off + (size_t)(lhi * 2) * SDIM + (n0 + lmod);

    v8f acc[2][4] = {};   // [mi][ni] 16x16 tiles

#pragma unroll 4
    for (int k = 0; k < CDIM; k += 4) {
        const float* ak = Abase + (size_t)k * SDIM;
        const float* bk = Bbase + (size_t)k * SDIM;

        v2f a[2], bf[4];
#pragma unroll
        for (int mi = 0; mi < 2; ++mi) {
            a[mi][0] = ak[mi * 16];
            a[mi][1] = ak[mi * 16 + SDIM];
        }
#pragma unroll
        for (int ni = 0; ni < 4; ++ni) {
            bf[ni][0] = bk[ni * 16];
            bf[ni][1] = bk[ni * 16 + SDIM];
        }

#pragma unroll
        for (int mi = 0; mi < 2; ++mi)
#pragma unroll
            for (int ni = 0; ni < 4; ++ni)
                acc[mi][ni] = __builtin_amdgcn_wmma_f32_16x16x4_f32(
                    /*neg_a=*/false, a[mi],
                    /*neg_b=*/false, bf[ni],
                    /*c_mod=*/(short)0, acc[mi][ni],
                    /*reuse_a=*/false, /*reuse_b=*/false);
    }

    // Epilogue: apply inverse norms + ReLU, store.
    const float* isrc = inv_src + (size_t)b * SDIM;
    const float* idst = inv_dst + (size_t)b * SDIM;
    float* obase = out + (size_t)b * SDIM * SDIM;

#pragma unroll
    for (int mi = 0; mi < 2; ++mi) {
        float im[8];
#pragma unroll
        for (int r = 0; r < 8; ++r)
            im[r] = isrc[m0 + mi * 16 + 8 * lhi + r];
#pragma unroll
        for (int ni = 0; ni < 4; ++ni) {
            const int n = n0 + ni * 16 + lmod;
            const float idn = idst[n];
#pragma unroll
            for (int r = 0; r < 8; ++r) {
                const int m = m0 + mi * 16 + 8 * lhi + r;
                float v = acc[mi][ni][r] * im[r] * idn;
                obase[(size_t)m * SDIM + n] = fmaxf(v, 0.0f);
            }
        }
    }
}

// ---------------------------------------------------------------------------
extern "C" void kernel_launch(void* const* d_in, const int* in_sizes, int n_in,
                              void* d_out, int out_size, void* d_ws, size_t ws_size,
                              hipStream_t stream) {
    const float* src = (const float*)d_in[0];   // [8, 256, 48, 48] fp32
    const float* dst = (const float*)d_in[1];   // [8, 256, 48, 48] fp32
    float* out = (float*)d_out;                 // [8, 2304, 2304] fp32

    float* inv = (float*)d_ws;                  // 2 * 8 * 2304 floats = 144 KiB
    float* inv_src = inv;
    float* inv_dst = inv + BATCH * SDIM;

    // Kernel 1: norms (2 * 8 * 2304 = 36864 threads)
    {
        int total = 2 * BATCH * SDIM;
        int blk = 256;
        int grid = (total + blk - 1) / blk;
        gcm_norms_kernel<<<grid, blk, 0, stream>>>(src, dst, inv);
    }

    // Kernel 2: WMMA GEMM + fused normalize + ReLU
    {
        dim3 grid(SDIM / 128, SDIM / 128, BATCH);   // (18, 18, 8)
        dim3 blk(256, 1, 1);
        gcm_wmma_kernel<<<grid, blk, 0, stream>>>(src, dst, inv_src, inv_dst, out);
    }
}